// GCN_63608465654253
// MI455X (gfx1250) — compile-verified
//
#include <hip/hip_runtime.h>
#include <math.h>

typedef __attribute__((ext_vector_type(2))) float v2f;
typedef __attribute__((ext_vector_type(8))) float v8f;

#define DIM_IN  128
#define DIM_H   256
#define DIM_OUT 128
#define BATCH   8
#define NPB     12500   /* nodes per batch */

// ---------------------------------------------------------------------------
// degree / normalization
// ---------------------------------------------------------------------------
__global__ void init_deg_kernel(float* __restrict__ deg, int n) {
    int i = blockIdx.x * blockDim.x + threadIdx.x;
    if (i < n) deg[i] = 1.0f;                    // self-loop contributes 1
}

__global__ void deg_edges_kernel(const int* __restrict__ ei, float* __restrict__ deg, int E) {
    int e = blockIdx.x * blockDim.x + threadIdx.x;
    if (e < E) atomicAdd(&deg[ei[E + e]], 1.0f); // dst row of edge_index
}

__global__ void make_dinv_kernel(float* __restrict__ deg, int n) {
    int i = blockIdx.x * blockDim.x + threadIdx.x;
    if (i < n) deg[i] = rsqrtf(deg[i]);          // deg >= 1 always (self-loop)
}

__global__ void zero_kernel(float* __restrict__ p, int count) {
    int i = blockIdx.x * blockDim.x + threadIdx.x;
    if (i < count) p[i] = 0.0f;
}

// ---------------------------------------------------------------------------
// GEMM  C[M,N] = A[M,K] @ W[K,N]   via V_WMMA_F32_16X16X4_F32
// block = 256 threads = 8 waves; one 16-row A tile in LDS shared by 8 waves.
// Each wave owns NT consecutive 16-col tiles (N == 8*16*NT, M % 16 == 0),
// so one A fragment feeds NT WMMAs and all stores are unconditional.
// ---------------------------------------------------------------------------
template <int K, int NT>
__global__ __launch_bounds__(256) void gemm_wmma_kernel(const float* __restrict__ A,
                                                        const float* __restrict__ W,
                                                        float* __restrict__ C,
                                                        int M, int N) {
    constexpr int KP = K + 4;                    // pad: kills 16-way LDS bank conflicts
    __shared__ float As[16 * KP];

    const int m0   = blockIdx.x * 16;
    const int wave = threadIdx.x >> 5;
    const int lane = threadIdx.x & 31;
    const int n0   = wave * (16 * NT);

    // cooperative, coalesced load of the 16 x K A tile (M % 16 == 0, exact grid)
    for (int idx = threadIdx.x; idx < 16 * K; idx += 256) {
        int r = idx / K, c = idx % K;
        As[r * KP + c] = A[(size_t)(m0 + r) * K + c];
    }
    __syncthreads();

    // A 16x4 f32 layout: lanes 0-15 -> M=lane, K={0,1}; lanes 16-31 -> K={2,3}
    const int r    = lane & 15;
    const int koff = (lane >> 4) << 1;
    const float* wcol = W + n0 + r;              // column base in W

    v8f acc[NT];
    #pragma unroll
    for (int t = 0; t < NT; ++t) acc[t] = (v8f){};

    #pragma unroll 4
    for (int k = 0; k < K; k += 4) {
        v2f a;
        a.x = As[r * KP + k + koff];
        a.y = As[r * KP + k + koff + 1];
        #pragma unroll
        for (int t = 0; t < NT; ++t) {
            v2f b;
            b.x = wcol[(size_t)(k + koff) * N + t * 16];
            b.y = wcol[(size_t)(k + koff + 1) * N + t * 16];
            acc[t] = __builtin_amdgcn_wmma_f32_16x16x4_f32(false, a, false, b,
                                                           (short)0, acc[t], false, false);
        }
    }

    // C/D layout: VGPR v -> M = v + 8*(lane>=16), N = lane&15 ; no bounds checks
    const int rowb = m0 + ((lane >> 4) << 3);
    #pragma unroll
    for (int t = 0; t < NT; ++t) {
        #pragma unroll
        for (int v = 0; v < 8; ++v) {
            C[(size_t)(rowb + v) * N + n0 + t * 16 + r] = acc[t][v];
        }
    }
}

// ---------------------------------------------------------------------------
// edge scatter:  agg[dst] += h[src] * dinv[src]*dinv[dst]
// one thread per (edge, 4 channels): float4 gather + 4 f32 atomics (L2-resident h)
// ---------------------------------------------------------------------------
template <int D>
__global__ void edge_agg_kernel(const int* __restrict__ ei, const float* __restrict__ h,
                                const float* __restrict__ dinv, float* __restrict__ agg,
                                int E) {
    constexpr int CPT = D / 4;
    int tid = blockIdx.x * blockDim.x + threadIdx.x;
    if (tid >= E * CPT) return;
    int e  = tid / CPT;
    int c4 = (tid % CPT) * 4;
    int s = ei[e], d = ei[E + e];
    float w = dinv[s] * dinv[d];
    float4 m = *(const float4*)(h + (size_t)s * D + c4);
    float* ap = agg + (size_t)d * D + c4;
    atomicAdd(ap + 0, m.x * w);
    atomicAdd(ap + 1, m.y * w);
    atomicAdd(ap + 2, m.z * w);
    atomicAdd(ap + 3, m.w * w);
}

// agg += self-loop message + bias (optionally ReLU), in place
template <int D, bool RELU>
__global__ void selfloop_bias_kernel(float* __restrict__ agg, const float* __restrict__ h,
                                     const float* __restrict__ dinv,
                                     const float* __restrict__ bias, int n) {
    int idx = blockIdx.x * blockDim.x + threadIdx.x;
    if (idx >= n * D) return;
    int i = idx / D, c = idx % D;
    float di = dinv[i];
    float v = agg[idx] + h[idx] * di * di + bias[c];
    agg[idx] = RELU ? fmaxf(v, 0.0f) : v;
}

// ---------------------------------------------------------------------------
// InstanceNorm1d stats: one block per batch, one thread per channel,
// coalesced 512B row reads, biased variance (torch semantics)
// ---------------------------------------------------------------------------
__global__ __launch_bounds__(128) void inorm_stats_kernel(const float* __restrict__ h,
                                                          float* __restrict__ stats) {
    int b = blockIdx.x, c = threadIdx.x;
    const float* base = h + (size_t)b * NPB * DIM_OUT + c;
    float s = 0.0f, sq = 0.0f;
    #pragma unroll 4
    for (int i = 0; i < NPB; ++i) {
        float v = base[(size_t)i * DIM_OUT];
        s += v; sq += v * v;
    }
    float mean = s * (1.0f / NPB);
    float var  = sq * (1.0f / NPB) - mean * mean;
    stats[b * DIM_OUT + c]                   = mean;
    stats[BATCH * DIM_OUT + b * DIM_OUT + c] = rsqrtf(var + 1e-5f);
}

// normalize (in place -> output #1) + log_softmax over 128 ch (-> output #2)
// one wave32 per node row, 4 channels per lane
__global__ __launch_bounds__(256) void norm_lsm_kernel(float* __restrict__ h,
                                                       float* __restrict__ lsm,
                                                       const float* __restrict__ stats,
                                                       const float* __restrict__ gamma,
                                                       const float* __restrict__ beta,
                                                       int n) {
    int wave = threadIdx.x >> 5, lane = threadIdx.x & 31;
    int row = blockIdx.x * 8 + wave;
    if (row >= n) return;
    int b = row / NPB;
    float* hrow = h + (size_t)row * DIM_OUT;

    float vals[4];
    float mx = -INFINITY;
    #pragma unroll
    for (int k = 0; k < 4; ++k) {
        int c = lane + 32 * k;
        float v = hrow[c];
        float mean = stats[b * DIM_OUT + c];
        float rstd = stats[BATCH * DIM_OUT + b * DIM_OUT + c];
        v = (v - mean) * rstd * gamma[c] + beta[c];
        vals[k] = v;
        hrow[c] = v;                              // output #1 in place
        mx = fmaxf(mx, v);
    }
    #pragma unroll
    for (int off = 16; off >= 1; off >>= 1)
        mx = fmaxf(mx, __shfl_xor(mx, off, 32));
    float se = 0.0f;
    #pragma unroll
    for (int k = 0; k < 4; ++k) se += expf(vals[k] - mx);
    #pragma unroll
    for (int off = 16; off >= 1; off >>= 1)
        se += __shfl_xor(se, off, 32);
    float lse = mx + logf(se);
    #pragma unroll
    for (int k = 0; k < 4; ++k)
        lsm[(size_t)row * DIM_OUT + lane + 32 * k] = vals[k] - lse;
}

// ---------------------------------------------------------------------------
extern "C" void kernel_launch(void* const* d_in, const int* in_sizes, int n_in,
                              void* d_out, int out_size, void* d_ws, size_t ws_size,
                              hipStream_t stream) {
    const float* x     = (const float*)d_in[0];
    const int*   ei    = (const int*)  d_in[1];
    const float* W1    = (const float*)d_in[2];
    const float* b1    = (const float*)d_in[3];
    const float* W2    = (const float*)d_in[4];
    const float* b2    = (const float*)d_in[5];
    const float* gamma = (const float*)d_in[6];
    const float* beta  = (const float*)d_in[7];

    const int n = in_sizes[0] / DIM_IN;   // 100000  (divisible by 16)
    const int E = in_sizes[1] / 2;        // 1,600,000

    // workspace layout (floats): dinv[n] | h1[n*256] (reused as h2[n*128]) |
    //                            agg1[n*256] | stats[2*B*128]
    float* ws    = (float*)d_ws;
    float* dinv  = ws;
    float* h1    = dinv + n;
    float* agg1  = h1 + (size_t)n * DIM_H;
    float* stats = agg1 + (size_t)n * DIM_H;
    float* h2    = h1;                    // h1 dead after layer-1 self-loop pass

    float* hout = (float*)d_out;                      // output #1 (also agg2 buf)
    float* lsm  = hout + (size_t)n * DIM_OUT;         // output #2

    const int T = 256;
    auto cdiv = [](int a, int b) { return (a + b - 1) / b; };

    // --- degrees -> dinv ---
    init_deg_kernel<<<cdiv(n, T), T, 0, stream>>>(dinv, n);
    deg_edges_kernel<<<cdiv(E, T), T, 0, stream>>>(ei, dinv, E);
    make_dinv_kernel<<<cdiv(n, T), T, 0, stream>>>(dinv, n);

    // --- layer 1: h1 = x @ W1 ; agg ; relu(agg + self + b1) ---
    gemm_wmma_kernel<DIM_IN, 2><<<n / 16, T, 0, stream>>>(x, W1, h1, n, DIM_H);
    zero_kernel<<<cdiv(n * DIM_H, T), T, 0, stream>>>(agg1, n * DIM_H);
    edge_agg_kernel<DIM_H><<<cdiv(E * (DIM_H / 4), T), T, 0, stream>>>(
        ei, h1, dinv, agg1, E);
    selfloop_bias_kernel<DIM_H, true><<<cdiv(n * DIM_H, T), T, 0, stream>>>(
        agg1, h1, dinv, b1, n);

    // --- layer 2: h2 = relu_out @ W2 ; agg into d_out ; + self + b2 ---
    gemm_wmma_kernel<DIM_H, 1><<<n / 16, T, 0, stream>>>(agg1, W2, h2, n, DIM_OUT);
    zero_kernel<<<cdiv(n * DIM_OUT, T), T, 0, stream>>>(hout, n * DIM_OUT);
    edge_agg_kernel<DIM_OUT><<<cdiv(E * (DIM_OUT / 4), T), T, 0, stream>>>(
        ei, h2, dinv, hout, E);
    selfloop_bias_kernel<DIM_OUT, false><<<cdiv(n * DIM_OUT, T), T, 0, stream>>>(
        hout, h2, dinv, b2, n);

    // --- InstanceNorm + log_softmax ---
    inorm_stats_kernel<<<BATCH, 128, 0, stream>>>(hout, stats);
    norm_lsm_kernel<<<cdiv(n, 8), T, 0, stream>>>(hout, lsm, stats, gamma, beta, n);
}